// DiffusionNetAutoencoder_25950192402638
// MI455X (gfx1250) — compile-verified
//
#include <hip/hip_runtime.h>

// ---------------------------------------------------------------------------
// ChebConv autoencoder on MI455X (gfx1250, wave32).
//   norm prep -> 4 layers of { K=6 Chebyshev terms: SpMM (atomic scatter) +
//   f32 WMMA GEMM accumulate, bias on k=0, ReLU on k=5 }.
// GEMMs: V_WMMA_F32_16X16X4_F32, 5 M-tiles per wave (one LDS B-fragment feeds
// 5 WMMAs), guard-free inner loop. SpMM: wave-per-edge float4 gathers +
// global_atomic_add_f32 into the L2-resident node array.
// ---------------------------------------------------------------------------

#define NODES 100000
#define EDGES 1600000
#define KCHEB 6
#define MT    5            // M-tiles per wave (100000 / (16*5) = 1250 exactly)

typedef float v2f __attribute__((ext_vector_type(2)));
typedef float v8f __attribute__((ext_vector_type(8)));

__device__ __forceinline__ void atomicAddF32(float* p, float v) {
    __hip_atomic_fetch_add(p, v, __ATOMIC_RELAXED, __HIP_MEMORY_SCOPE_AGENT);
}

// -------------------------------- utility ----------------------------------

__global__ void fill_kernel(float* __restrict__ dst, float val, int n) {
    int i = blockIdx.x * blockDim.x + threadIdx.x;
    if (i < n) dst[i] = val;
}

__global__ void scale_copy_kernel(float* __restrict__ dst,
                                  const float* __restrict__ src,
                                  float s, int n) {
    int i = blockIdx.x * blockDim.x + threadIdx.x;
    if (i < n) dst[i] = s * src[i];
}

// --------------------------- edge normalization ----------------------------

__global__ void edge_deg_kernel(const int* __restrict__ row,
                                const int* __restrict__ col,
                                const float* __restrict__ w,
                                float* __restrict__ deg, int E) {
    int e = blockIdx.x * blockDim.x + threadIdx.x;
    if (e >= E) return;
    int r = row[e];
    float v = (r == col[e]) ? 0.f : w[e];
    if (v != 0.f) atomicAddF32(&deg[r], v);
}

__global__ void dis_kernel(float* __restrict__ deg, int n) {
    int i = blockIdx.x * blockDim.x + threadIdx.x;
    if (i < n) {
        float d = deg[i];
        deg[i] = (d > 0.f) ? rsqrtf(fmaxf(d, 1e-30f)) : 0.f;
    }
}

__global__ void norm_kernel(const int* __restrict__ row,
                            const int* __restrict__ col,
                            const float* __restrict__ w,
                            const float* __restrict__ dis,
                            float* __restrict__ nrm, int E) {
    int e = blockIdx.x * blockDim.x + threadIdx.x;
    if (e >= E) return;
    int r = row[e], c = col[e];
    float v = (r == c) ? 0.f : w[e];
    nrm[e] = -dis[r] * v * dis[c];
}

// ------------------------------- SpMM scatter ------------------------------
// One wave32 per edge; lanes cover the feature dim. b128 gathers when F%4==0;
// scatters via f32 HW atomics (targets are L2-resident: <= 51 MB vs 192 MB L2).

__global__ void spmm_kernel(const int* __restrict__ row,
                            const int* __restrict__ col,
                            const float* __restrict__ nrm,
                            const float* __restrict__ h,
                            float* __restrict__ out,
                            int E, int F, float scale) {
    int gid  = blockIdx.x * blockDim.x + threadIdx.x;
    int wid  = gid >> 5;
    int lane = gid & 31;
    if (wid >= E) return;                        // wave-uniform
    float s = scale * nrm[wid];
    if (s == 0.f) return;                        // wave-uniform
    int r = row[wid], c = col[wid];
    const float* hs = h + (size_t)c * F;
    float*       os = out + (size_t)r * F;
    if ((F & 3) == 0) {
        for (int f = lane * 4; f < F; f += 128) {
            float4 hv = *(const float4*)(hs + f);
            atomicAddF32(os + f + 0, s * hv.x);
            atomicAddF32(os + f + 1, s * hv.y);
            atomicAddF32(os + f + 2, s * hv.z);
            atomicAddF32(os + f + 3, s * hv.w);
        }
    } else {
        for (int f = lane; f < F; f += 32)
            atomicAddF32(&os[f], s * hs[f]);
    }
}

// ------------------------------ WMMA f32 GEMM ------------------------------
// out[M,N] (+)= A[M,Kd] * W[Kd,N]   (flags bit0: init C from bias;
//                                    flags bit1: ReLU on store)
// 8 waves / workgroup; each wave owns MT=5 M-tiles x 1 N-tile. W panel is
// zero-padded to Kpad x Npad in LDS, so A over-reads past a row tail multiply
// zeros -> inner loop needs NO bounds checks (callers guarantee the A
// allocation extends past row M-1 by >= Kpad-Kd elements).
// VGPR layouts per CDNA5 ISA 7.12.2:
//   A 16x4  f32: lane%16 = M row; lane/16 selects K pair {0,1}|{2,3}.
//   B 4x16  f32: lane%16 = N col; lane/16 selects K rows {0,1}|{2,3}.
//   C 16x16 f32: vgpr v -> row v (lanes 0-15) / row v+8 (lanes 16-31).

#define GEMM_SMEM 4096  // floats; max Kpad*Npad = 128*32

__global__ __launch_bounds__(256) void cheb_gemm_kernel(
        const float* __restrict__ A, const float* __restrict__ W,
        const float* __restrict__ bias, float* __restrict__ out,
        int M, int Kd, int N, int flags) {
    __shared__ float sB[GEMM_SMEM];
    const int Kpad = (Kd + 3) & ~3;
    const int Npad = (N + 15) & ~15;

    for (int idx = threadIdx.x; idx < Kpad * Npad; idx += 256) {
        int kk = idx / Npad, nn = idx % Npad;
        sB[idx] = (kk < Kd && nn < N) ? W[kk * N + nn] : 0.f;
    }
    __syncthreads();

    const int wave    = threadIdx.x >> 5;
    const int lane    = threadIdx.x & 31;
    const int Ntiles  = Npad >> 4;
    const int Mgroups = M / (16 * MT);           // 1250 (exact by construction)
    const int group   = blockIdx.x * 8 + wave;
    if (group >= Mgroups * Ntiles) return;       // wave-uniform: EXEC all-1s

    const int mGrp = group / Ntiles;
    const int nTile = group % Ntiles;
    const int l16  = lane & 15;
    const int half = lane >> 4;                  // 0 | 1
    const int n    = nTile * 16 + l16;           // B/C column for this lane
    const int m0   = mGrp * (16 * MT);

    const float* Ar[MT];
#pragma unroll
    for (int t = 0; t < MT; ++t) {
        Ar[t] = A + (size_t)(m0 + t * 16 + l16) * Kd;
        __builtin_prefetch(Ar[t], 0, 3);         // global_prefetch_b8
    }

    v8f c[MT];
    if (flags & 1) {                             // init from bias
        float bv = (n < N) ? bias[n] : 0.f;
#pragma unroll
        for (int t = 0; t < MT; ++t)
#pragma unroll
            for (int v = 0; v < 8; ++v) c[t][v] = bv;
    } else if (N == Npad) {                      // accumulate: reload C tiles
#pragma unroll
        for (int t = 0; t < MT; ++t)
#pragma unroll
            for (int v = 0; v < 8; ++v)
                c[t][v] = out[(size_t)(m0 + t * 16 + half * 8 + v) * N + n];
    } else {
#pragma unroll
        for (int t = 0; t < MT; ++t)
#pragma unroll
            for (int v = 0; v < 8; ++v)
                c[t][v] = (n < N)
                    ? out[(size_t)(m0 + t * 16 + half * 8 + v) * N + n] : 0.f;
    }

    if ((Kd & 1) == 0) {                         // 8B-aligned rows -> b64 loads
        for (int k0 = 0; k0 < Kpad; k0 += 4) {
            int ka = k0 + half * 2;
            v2f b;
            b.x = sB[ka * Npad + n - (nTile ? 0 : 0)];  // n == nl (padded col)
            b.y = sB[(ka + 1) * Npad + n];
#pragma unroll
            for (int t = 0; t < MT; ++t) {
                v2f a = *(const v2f*)(Ar[t] + ka);
                c[t] = __builtin_amdgcn_wmma_f32_16x16x4_f32(
                        false, a, false, b, (short)0, c[t], false, false);
            }
        }
    } else {                                     // odd Kd (=3): scalar A loads
        for (int k0 = 0; k0 < Kpad; k0 += 4) {
            int ka = k0 + half * 2;
            v2f b;
            b.x = sB[ka * Npad + n];
            b.y = sB[(ka + 1) * Npad + n];
#pragma unroll
            for (int t = 0; t < MT; ++t) {
                v2f a;
                a.x = Ar[t][ka];
                a.y = Ar[t][ka + 1];
                c[t] = __builtin_amdgcn_wmma_f32_16x16x4_f32(
                        false, a, false, b, (short)0, c[t], false, false);
            }
        }
    }

    const bool relu = (flags & 2) != 0;
    if (N == Npad) {
#pragma unroll
        for (int t = 0; t < MT; ++t)
#pragma unroll
            for (int v = 0; v < 8; ++v) {
                float val = c[t][v];
                if (relu) val = fmaxf(val, 0.f);
                out[(size_t)(m0 + t * 16 + half * 8 + v) * N + n] = val;
            }
    } else {
#pragma unroll
        for (int t = 0; t < MT; ++t)
#pragma unroll
            for (int v = 0; v < 8; ++v) {
                if (n < N) {
                    float val = c[t][v];
                    if (relu) val = fmaxf(val, 0.f);
                    out[(size_t)(m0 + t * 16 + half * 8 + v) * N + n] = val;
                }
            }
    }
}

// --------------------------------- driver ----------------------------------

extern "C" void kernel_launch(void* const* d_in, const int* in_sizes, int n_in,
                              void* d_out, int out_size, void* d_ws, size_t ws_size,
                              hipStream_t stream) {
    const float* x   = (const float*)d_in[0];
    const int*   ei  = (const int*)d_in[1];   // [2, E] int32
    const float* lap = (const float*)d_in[2];
    const float* Wl[4] = {(const float*)d_in[3], (const float*)d_in[5],
                          (const float*)d_in[7], (const float*)d_in[9]};
    const float* bl[4] = {(const float*)d_in[4], (const float*)d_in[6],
                          (const float*)d_in[8], (const float*)d_in[10]};
    const int* row = ei;
    const int* col = ei + EDGES;

    float* ws   = (float*)d_ws;
    float* deg  = ws;  ws += NODES;                 // becomes dis in-place
    float* nrm  = ws;  ws += EDGES;
    float* buf[3];
    buf[0] = ws;  ws += (size_t)NODES * 128;
    buf[1] = ws;  ws += (size_t)NODES * 128;
    buf[2] = ws;  ws += (size_t)NODES * 128;
    float* act1 = ws;  ws += (size_t)NODES * 128;
    float* act2 = ws;  ws += (size_t)NODES * 32;
    float* act3 = ws;  ws += (size_t)NODES * 128;   // also holds the x copy

    const int TB = 256;
    const int nodeBlocks = (NODES + TB - 1) / TB;
    const int edgeBlocks = (EDGES + TB - 1) / TB;
    const int spmmBlocks = EDGES / 8;               // one wave32 per edge

    // ---- edge normalization: deg -> dis -> norm ----
    fill_kernel<<<nodeBlocks, TB, 0, stream>>>(deg, 0.f, NODES);
    edge_deg_kernel<<<edgeBlocks, TB, 0, stream>>>(row, col, lap, deg, EDGES);
    dis_kernel<<<nodeBlocks, TB, 0, stream>>>(deg, NODES);
    norm_kernel<<<edgeBlocks, TB, 0, stream>>>(row, col, lap, deg, nrm, EDGES);

    // x must live in an allocation with slack past the last row (Kpad over-
    // read) and must not be mutated -> stage it once into act3.
    scale_copy_kernel<<<(NODES * 3 + TB - 1) / TB, TB, 0, stream>>>(
        act3, x, 1.f, NODES * 3);

    // ---- one ChebConv + ReLU layer (in is read-only; 3 rotating scratch) ----
    auto run_layer = [&](const float* in, int Fin, float* outp, int Fout,
                         const float* W, const float* bias) {
        const int nIn   = NODES * Fin;
        const int cpBlk = (nIn + TB - 1) / TB;
        const int Npad  = (Fout + 15) & ~15;
        const int grps  = (NODES / (16 * MT)) * (Npad >> 4);
        const int gBlk  = (grps + 7) / 8;
        const size_t Wsz = (size_t)Fin * Fout;

        const float* t0 = in;                    // Tx0
        float* t1 = buf[0];                      // Tx1
        int nxt = 1;                             // next scratch index

        // out = bias + Tx0 @ W[0]
        cheb_gemm_kernel<<<gBlk, TB, 0, stream>>>(t0, W, bias, outp,
                                                  NODES, Fin, Fout, /*init*/1);
        // Tx1 = L_hat @ Tx0
        fill_kernel<<<cpBlk, TB, 0, stream>>>(t1, 0.f, nIn);
        spmm_kernel<<<spmmBlocks, TB, 0, stream>>>(row, col, nrm, t0, t1,
                                                   EDGES, Fin, 1.f);
        cheb_gemm_kernel<<<gBlk, TB, 0, stream>>>(t1, W + Wsz, bias, outp,
                                                  NODES, Fin, Fout, 0);
        // Tx_k = 2 L_hat Tx_{k-1} - Tx_{k-2}
        for (int k = 2; k < KCHEB; ++k) {
            float* t2 = buf[nxt];
            nxt = (nxt + 1) % 3;
            scale_copy_kernel<<<cpBlk, TB, 0, stream>>>(t2, (float*)t0, -1.f, nIn);
            spmm_kernel<<<spmmBlocks, TB, 0, stream>>>(row, col, nrm, t1, t2,
                                                       EDGES, Fin, 2.f);
            cheb_gemm_kernel<<<gBlk, TB, 0, stream>>>(t2, W + (size_t)k * Wsz,
                                                      bias, outp, NODES, Fin,
                                                      Fout,
                                                      (k == KCHEB - 1) ? 2 : 0);
            t0 = t1; t1 = t2;
        }
    };

    run_layer(act3, 3,   act1, 128, Wl[0], bl[0]);   // encoder (act3 = x copy)
    run_layer(act1, 128, act2, 32,  Wl[1], bl[1]);
    run_layer(act2, 32,  act3, 128, Wl[2], bl[2]);   // decoder (overwrites x copy)
    run_layer(act3, 128, (float*)d_out, 3, Wl[3], bl[3]);
}